// WienerPostFilter_23948737642983
// MI455X (gfx1250) — compile-verified
//
#include <hip/hip_runtime.h>
#include <math.h>

typedef __attribute__((ext_vector_type(2))) float v2f;
typedef __attribute__((ext_vector_type(8))) float v8f;

#define NFFT        512
#define HOP         128
#define KBINS       257            // NFFT/2 + 1
#define NCOL        514            // re (257) + im (257) columns
#define SPEC_STRIDE 528            // 33 * 16, padded row stride of spectrum
#define NT_FWD      33             // ceil(514/16) output col-tiles, forward
#define NT_INV      32             // 512/16 output col-tiles, inverse
#define KSTEP_FWD   128            // 512/4 k-steps, forward
#define KSTEP_INV   129            // ceil(516/4) k-steps, inverse
#define CHUNKS      512
#define DECAYF      0.999f
#define TWO_PIF     6.28318530717958647692f
#define LDSA_STRIDE 516            // 516 % 64 == 4 -> conflict-free A-frag b64 reads

#define WMMA_F32X4(a, b, c) \
    __builtin_amdgcn_wmma_f32_16x16x4_f32(false, (a), false, (b), (short)0, (c), false, false)

// ---------------------------------------------------------------------------
// Setup: build forward DFT basis (window folded in), packed in WMMA-B
// fragment order:  elem index = ((s*NT + nt)*32 + lane)*2 + h
// lane<16 : N = nt*16 + lane,     K = 4s + h
// lane>=16: N = nt*16 + lane-16,  K = 4s + 2 + h
// ---------------------------------------------------------------------------
__global__ void build_bfwd(float* __restrict__ Bf) {
    int idx = blockIdx.x * blockDim.x + threadIdx.x;
    const int total = KSTEP_FWD * NT_FWD * 64;
    if (idx >= total) return;
    int h    = idx & 1;
    int lane = (idx >> 1) & 31;
    int snt  = idx >> 6;
    int nt   = snt % NT_FWD;
    int s    = snt / NT_FWD;
    int n = 4 * s + ((lane < 16) ? h : 2 + h);   // time index (K dim)
    int j = nt * 16 + (lane & 15);               // output column (freq re/im)
    float v = 0.0f;
    if (j < NCOL) {
        int k = (j < KBINS) ? j : (j - KBINS);
        int ph = (k * n) & (NFFT - 1);           // exact integer phase reduction
        float ang = TWO_PIF * (float)ph / (float)NFFT;
        float sv, cv; sincosf(ang, &sv, &cv);
        float w = 0.5f * (1.0f - cosf(TWO_PIF * (float)n / (float)NFFT));
        v = (j < KBINS) ? (w * cv) : (-w * sv);  // re: cos, im: -sin
    }
    Bf[idx] = v;
}

// Inverse basis: y[n] = sum_k (c_k/N)*(Re_k cos - Im_k sin) * win[n]
__global__ void build_binv(float* __restrict__ Bi) {
    int idx = blockIdx.x * blockDim.x + threadIdx.x;
    const int total = KSTEP_INV * NT_INV * 64;
    if (idx >= total) return;
    int h    = idx & 1;
    int lane = (idx >> 1) & 31;
    int snt  = idx >> 6;
    int nt   = snt % NT_INV;
    int s    = snt / NT_INV;
    int j = 4 * s + ((lane < 16) ? h : 2 + h);   // K dim = spectrum column 0..515
    int n = nt * 16 + (lane & 15);               // output column = time 0..511
    float v = 0.0f;
    if (j < NCOL) {
        int k = (j < KBINS) ? j : (j - KBINS);
        float c = (k == 0 || k == NFFT / 2) ? 1.0f : 2.0f;
        int ph = (k * n) & (NFFT - 1);
        float ang = TWO_PIF * (float)ph / (float)NFFT;
        float sv, cv; sincosf(ang, &sv, &cv);
        float w = 0.5f * (1.0f - cosf(TWO_PIF * (float)n / (float)NFFT));
        float sc = (c / (float)NFFT) * w;
        v = (j < KBINS) ? (sc * cv) : (-sc * sv);
    }
    Bi[idx] = v;
}

__global__ void build_win2(float* __restrict__ win2) {
    int n = blockIdx.x * blockDim.x + threadIdx.x;
    if (n >= NFFT) return;
    float w = 0.5f * (1.0f - cosf(TWO_PIF * (float)n / (float)NFFT));
    win2[n] = w * w;
}

// ---------------------------------------------------------------------------
// Forward STFT GEMM: spec[F,514] = frames[F,512] x Bfwd[512,514]
// One block = one 16-frame M-tile; frames gathered (reflect pad) into LDS.
// Each wave runs TWO N-tiles concurrently: one ds_load_b64 feeds two WMMAs,
// and the two accumulator chains are independent.
// ---------------------------------------------------------------------------
__global__ __launch_bounds__(256)
void stft_gemm(const float* __restrict__ x, const float* __restrict__ Bf,
               float* __restrict__ spec, int L, int F) {
    __shared__ float sA[16 * LDSA_STRIDE];
    const int f0   = blockIdx.x * 16;
    const int base = f0 * HOP - (NFFT / 2);      // padded -> signal offset
    for (int i = threadIdx.x; i < 16 * NFFT; i += 256) {
        int m = i >> 9, n = i & (NFFT - 1);
        int p = base + m * HOP + n;
        if (p < 0) p = -p;                        // reflect left
        else if (p >= L) p = 2 * L - 2 - p;       // reflect right (covers tail tile)
        sA[m * LDSA_STRIDE + n] = x[p];
    }
    __syncthreads();

    const int lane  = threadIdx.x & 31;
    const int wv    = threadIdx.x >> 5;
    const int mrow  = lane & 15;
    const int aoff  = mrow * LDSA_STRIDE + ((lane < 16) ? 0 : 2);
    const int crow0 = f0 + 8 * (lane >> 4);
    const int ccol  = lane & 15;

    for (int j = 0; wv + 8 * j < NT_FWD; ) {
        const int ntA = wv + 8 * j;
        const int ntB = ntA + 8;
        if (ntB < NT_FWD) {
            v8f acc0 = {0.f, 0.f, 0.f, 0.f, 0.f, 0.f, 0.f, 0.f};
            v8f acc1 = {0.f, 0.f, 0.f, 0.f, 0.f, 0.f, 0.f, 0.f};
            const v2f* b0 = (const v2f*)Bf + (ntA * 32 + lane);
            const v2f* b1 = (const v2f*)Bf + (ntB * 32 + lane);
            for (int s = 0; s < KSTEP_FWD; ++s) {
                v2f a  = *(const v2f*)(&sA[aoff + 4 * s]);
                v2f v0 = b0[(size_t)s * (NT_FWD * 32)];
                v2f v1 = b1[(size_t)s * (NT_FWD * 32)];
                acc0 = WMMA_F32X4(a, v0, acc0);
                acc1 = WMMA_F32X4(a, v1, acc1);
            }
            const long colA = (long)ntA * 16 + ccol;
            const long colB = (long)ntB * 16 + ccol;
            #pragma unroll
            for (int v = 0; v < 8; ++v) {
                spec[(long)(crow0 + v) * SPEC_STRIDE + colA] = acc0[v];
                spec[(long)(crow0 + v) * SPEC_STRIDE + colB] = acc1[v];
            }
            j += 2;
        } else {
            v8f acc = {0.f, 0.f, 0.f, 0.f, 0.f, 0.f, 0.f, 0.f};
            const v2f* b0 = (const v2f*)Bf + (ntA * 32 + lane);
            for (int s = 0; s < KSTEP_FWD; ++s) {
                v2f a  = *(const v2f*)(&sA[aoff + 4 * s]);
                v2f v0 = b0[(size_t)s * (NT_FWD * 32)];
                acc = WMMA_F32X4(a, v0, acc);
            }
            const long colA = (long)ntA * 16 + ccol;
            #pragma unroll
            for (int v = 0; v < 8; ++v)
                spec[(long)(crow0 + v) * SPEC_STRIDE + colA] = acc[v];
            j += 1;
        }
    }
}

// ---------------------------------------------------------------------------
// Scan pass A: per (freq, chunk) composed running-max with zero seed.
// ---------------------------------------------------------------------------
__global__ void scan_chunk_reduce(const float* __restrict__ spec,
                                  float* __restrict__ chunkM, int F, int CL) {
    int k = blockIdx.x * blockDim.x + threadIdx.x;
    int c = blockIdx.y;
    if (k >= KBINS) return;
    int fs = c * CL;
    int fe = fs + CL; if (fe > F) fe = F;
    float r = 0.0f;
    for (int f = fs; f < fe; ++f) {
        const float* row = spec + (long)f * SPEC_STRIDE;
        float re = row[k], im = row[KBINS + k];
        float m = sqrtf(re * re + im * im);
        r = fmaxf(DECAYF * r, m);
    }
    chunkM[(long)k * CHUNKS + c] = r;
}

// Scan pass B: serial combine over 512 chunks per frequency (exclusive r_in).
__global__ void scan_exclusive(const float* __restrict__ chunkM,
                               float* __restrict__ rIn, int F, int CL) {
    int k = blockIdx.x * blockDim.x + threadIdx.x;
    if (k >= KBINS) return;
    float r = 0.0f;
    for (int c = 0; c < CHUNKS; ++c) {
        rIn[(long)k * CHUNKS + c] = r;
        int fs = c * CL;
        int fe = fs + CL; if (fe > F) fe = F;
        int len = (fe > fs) ? (fe - fs) : 0;
        float dp = __powf(DECAYF, (float)len);   // r_out = max(r_in*d^len, Mc)
        r = fmaxf(r * dp, chunkM[(long)k * CHUNKS + c]);
    }
}

// Scan pass C: replay each chunk with correct r_in; scale spec in place by
// Wiener gain g = snr/(1+snr), snr = m^2 / max(0.5 * rmax^2, 1e-8).
__global__ void scan_apply(float* __restrict__ spec,
                           const float* __restrict__ rIn, int F, int CL) {
    int k = blockIdx.x * blockDim.x + threadIdx.x;
    int c = blockIdx.y;
    if (k >= KBINS) return;
    int fs = c * CL;
    int fe = fs + CL; if (fe > F) fe = F;
    float r = rIn[(long)k * CHUNKS + c];
    for (int f = fs; f < fe; ++f) {
        float* row = spec + (long)f * SPEC_STRIDE;
        float re = row[k], im = row[KBINS + k];
        float m = sqrtf(re * re + im * im);
        r = fmaxf(DECAYF * r, m);
        float snr = (m * m) / fmaxf(0.5f * (r * r), 1e-8f);
        float g = snr / (1.0f + snr);
        row[k] = re * g;
        row[KBINS + k] = im * g;
    }
}

// ---------------------------------------------------------------------------
// Inverse GEMM: yframes[F,512] = spec[F,514] x Binv[514,512]
// spec cols 514..527 are exactly zero (forward B columns were zero), so the
// 516-wide LDS stage needs no bounds logic. NT_INV = 32 = 8 waves * 2 pairs,
// so every wave runs exactly two dual-accumulator passes.
// ---------------------------------------------------------------------------
__global__ __launch_bounds__(256)
void istft_gemm(const float* __restrict__ spec, const float* __restrict__ Bi,
                float* __restrict__ yf) {
    __shared__ float sA[16 * LDSA_STRIDE];
    const long f0 = (long)blockIdx.x * 16;
    for (int i = threadIdx.x; i < 16 * LDSA_STRIDE; i += 256) {
        int m = i / LDSA_STRIDE, j = i % LDSA_STRIDE;
        sA[i] = spec[(f0 + m) * SPEC_STRIDE + j];
    }
    __syncthreads();

    const int lane  = threadIdx.x & 31;
    const int wv    = threadIdx.x >> 5;
    const int mrow  = lane & 15;
    const int aoff  = mrow * LDSA_STRIDE + ((lane < 16) ? 0 : 2);
    const int crow0 = 8 * (lane >> 4);
    const int ccol  = lane & 15;

    #pragma unroll
    for (int half = 0; half < 2; ++half) {
        const int ntA = wv + 16 * half;
        const int ntB = ntA + 8;
        v8f acc0 = {0.f, 0.f, 0.f, 0.f, 0.f, 0.f, 0.f, 0.f};
        v8f acc1 = {0.f, 0.f, 0.f, 0.f, 0.f, 0.f, 0.f, 0.f};
        const v2f* b0 = (const v2f*)Bi + (ntA * 32 + lane);
        const v2f* b1 = (const v2f*)Bi + (ntB * 32 + lane);
        for (int s = 0; s < KSTEP_INV; ++s) {            // k up to 515 in LDS
            v2f a  = *(const v2f*)(&sA[aoff + 4 * s]);
            v2f v0 = b0[(size_t)s * (NT_INV * 32)];
            v2f v1 = b1[(size_t)s * (NT_INV * 32)];
            acc0 = WMMA_F32X4(a, v0, acc0);
            acc1 = WMMA_F32X4(a, v1, acc1);
        }
        const long colA = (long)ntA * 16 + ccol;
        const long colB = (long)ntB * 16 + ccol;
        #pragma unroll
        for (int v = 0; v < 8; ++v) {
            yf[(f0 + crow0 + v) * NFFT + colA] = acc0[v];
            yf[(f0 + crow0 + v) * NFFT + colB] = acc1[v];
        }
    }
}

// ---------------------------------------------------------------------------
// Gather-style overlap-add + window^2 normalization (deterministic, no atomics)
// ---------------------------------------------------------------------------
__global__ void ola(const float* __restrict__ yf, const float* __restrict__ win2,
                    float* __restrict__ out, int L, int F) {
    int t = blockIdx.x * blockDim.x + threadIdx.x;
    if (t >= L) return;
    int tp = t + NFFT / 2;
    int fhi = tp >> 7; if (fhi > F - 1) fhi = F - 1;
    int a = tp - (NFFT - 1);
    int flo = (a > 0) ? ((a + HOP - 1) >> 7) : 0;
    float acc = 0.0f, ws = 0.0f;
    for (int f = flo; f <= fhi; ++f) {
        int n = tp - (f << 7);
        acc += yf[(long)f * NFFT + n];
        ws  += win2[n];
    }
    out[t] = acc / fmaxf(ws, 1e-11f);
}

// ---------------------------------------------------------------------------
extern "C" void kernel_launch(void* const* d_in, const int* in_sizes, int n_in,
                              void* d_out, int out_size, void* d_ws, size_t ws_size,
                              hipStream_t stream) {
    const float* x = (const float*)d_in[0];
    float* out = (float*)d_out;
    const int L = in_sizes[0];                    // 8388608
    const int F = 1 + L / HOP;                    // 65537 frames
    const int Mtiles = (F + 15) / 16;             // 4097
    const long Fp = (long)Mtiles * 16;
    const int CL = (F + CHUNKS - 1) / CHUNKS;     // 129

    float* ws = (float*)d_ws;
    size_t o = 0;
    float* Bf     = ws + o; o += (size_t)KSTEP_FWD * NT_FWD * 64;
    float* Bi     = ws + o; o += (size_t)KSTEP_INV * NT_INV * 64;
    float* win2   = ws + o; o += NFFT;
    o = (o + 15) & ~(size_t)15;
    float* spec   = ws + o; o += (size_t)Fp * SPEC_STRIDE;
    float* yf     = ws + o; o += (size_t)Fp * NFFT;
    float* chunkM = ws + o; o += (size_t)KBINS * CHUNKS;
    float* rIn    = ws + o; o += (size_t)KBINS * CHUNKS;
    (void)ws_size; (void)n_in; (void)out_size;

    build_bfwd<<<(KSTEP_FWD * NT_FWD * 64 + 255) / 256, 256, 0, stream>>>(Bf);
    build_binv<<<(KSTEP_INV * NT_INV * 64 + 255) / 256, 256, 0, stream>>>(Bi);
    build_win2<<<(NFFT + 255) / 256, 256, 0, stream>>>(win2);

    stft_gemm<<<Mtiles, 256, 0, stream>>>(x, Bf, spec, L, F);

    dim3 sg((KBINS + 63) / 64, CHUNKS);
    scan_chunk_reduce<<<sg, 64, 0, stream>>>(spec, chunkM, F, CL);
    scan_exclusive<<<(KBINS + 63) / 64, 64, 0, stream>>>(chunkM, rIn, F, CL);
    scan_apply<<<sg, 64, 0, stream>>>(spec, rIn, F, CL);

    istft_gemm<<<Mtiles, 256, 0, stream>>>(spec, Bi, yf);
    ola<<<(L + 255) / 256, 256, 0, stream>>>(yf, win2, out, L, F);
}